// QRNN_18622978195481
// MI455X (gfx1250) — compile-verified
//
#include <hip/hip_runtime.h>
#include <hip/hip_bf16.h>

#define WIRES 8
#define TSTEPS 256
#define BATCH 256
#define FEAT 64
#define UNITS 128
#define BT (BATCH * TSTEPS)   // 65536 rows

typedef __attribute__((ext_vector_type(2))) float v2f;
typedef __attribute__((ext_vector_type(8))) float v8f;

// ---------------------------------------------------------------------------
// Quantum gate primitives. Statevector: 256 complex amps per wave32.
// Amp index i = (reg<<5) | lane.  Reference wire w lives at bit (7-w).
// Bits 0..4 -> lane bits (cross-lane via shfl_xor), bits 5..7 -> reg bits.
// ---------------------------------------------------------------------------

template<int BP>
__device__ __forceinline__ void g_ry(float (&re)[8], float (&im)[8], float c, float s) {
  if constexpr (BP < 5) {
    const int m = 1 << BP;
    const float sg = (threadIdx.x & m) ? s : -s;   // bit0: c*a0 - s*a1 ; bit1: c*a1 + s*a0
#pragma unroll
    for (int r = 0; r < 8; ++r) {
      float pr = __shfl_xor(re[r], m, 32);
      float pi = __shfl_xor(im[r], m, 32);
      re[r] = c * re[r] + sg * pr;
      im[r] = c * im[r] + sg * pi;
    }
  } else {
    constexpr int m = 1 << (BP - 5);
#pragma unroll
    for (int r = 0; r < 8; ++r) {
      if (!(r & m)) {
        const int r1 = r | m;
        float a0r = re[r], a0i = im[r], a1r = re[r1], a1i = im[r1];
        re[r]  = c * a0r - s * a1r;  im[r]  = c * a0i - s * a1i;
        re[r1] = s * a0r + c * a1r;  im[r1] = s * a0i + c * a1i;
      }
    }
  }
}

template<int BP>
__device__ __forceinline__ void g_rx(float (&re)[8], float (&im)[8], float c, float s) {
  // RX: new = c*self - i*s*partner  -> re' = c*re + s*partner_im ; im' = c*im - s*partner_re
  if constexpr (BP < 5) {
    const int m = 1 << BP;
#pragma unroll
    for (int r = 0; r < 8; ++r) {
      float pr = __shfl_xor(re[r], m, 32);
      float pi = __shfl_xor(im[r], m, 32);
      re[r] = c * re[r] + s * pi;
      im[r] = c * im[r] - s * pr;
    }
  } else {
    constexpr int m = 1 << (BP - 5);
#pragma unroll
    for (int r = 0; r < 8; ++r) {
      if (!(r & m)) {
        const int r1 = r | m;
        float a0r = re[r], a0i = im[r], a1r = re[r1], a1i = im[r1];
        re[r]  = c * a0r + s * a1i;  im[r]  = c * a0i - s * a1r;
        re[r1] = c * a1r + s * a0i;  im[r1] = c * a1i - s * a0r;
      }
    }
  }
}

template<int BP>
__device__ __forceinline__ void g_rz(float (&re)[8], float (&im)[8], float c, float s) {
  // bit0: *(c - i s)  -> re' = c*re + s*im ; im' = c*im - s*re
  // bit1: *(c + i s)  -> re' = c*re - s*im ; im' = c*im + s*re
  if constexpr (BP < 5) {
    const int m = 1 << BP;
    const float sg = (threadIdx.x & m) ? -s : s;
#pragma unroll
    for (int r = 0; r < 8; ++r) {
      float rr = re[r], ii = im[r];
      re[r] = c * rr + sg * ii;
      im[r] = c * ii - sg * rr;
    }
  } else {
    constexpr int m = 1 << (BP - 5);
#pragma unroll
    for (int r = 0; r < 8; ++r) {
      const float sg = (r & m) ? -s : s;
      float rr = re[r], ii = im[r];
      re[r] = c * rr + sg * ii;
      im[r] = c * ii - sg * rr;
    }
  }
}

// CNOT with control bit CB, target bit TB: new[i] = (bit CB of i) ? old[i ^ (1<<TB)] : old[i]
template<int CB, int TB>
__device__ __forceinline__ void cnot(float (&re)[8], float (&im)[8]) {
  if constexpr (TB < 5) {
    const int tm = 1 << TB;
    if constexpr (CB < 5) {
      const bool c = (threadIdx.x & (1 << CB)) != 0;
#pragma unroll
      for (int r = 0; r < 8; ++r) {
        float pr = __shfl_xor(re[r], tm, 32);
        float pi = __shfl_xor(im[r], tm, 32);
        re[r] = c ? pr : re[r];
        im[r] = c ? pi : im[r];
      }
    } else {
      constexpr int cm = 1 << (CB - 5);
#pragma unroll
      for (int r = 0; r < 8; ++r) {
        if (r & cm) {
          re[r] = __shfl_xor(re[r], tm, 32);
          im[r] = __shfl_xor(im[r], tm, 32);
        }
      }
    }
  } else {
    constexpr int tm = 1 << (TB - 5);
    if constexpr (CB < 5) {
      const bool c = (threadIdx.x & (1 << CB)) != 0;
#pragma unroll
      for (int r = 0; r < 8; ++r) {
        if (!(r & tm)) {
          const int r1 = r | tm;
          float t0r = re[r], t0i = im[r];
          re[r]  = c ? re[r1] : re[r];   im[r]  = c ? im[r1] : im[r];
          re[r1] = c ? t0r    : re[r1];  im[r1] = c ? t0i    : im[r1];
        }
      }
    } else {
      constexpr int cm = 1 << (CB - 5);
#pragma unroll
      for (int r = 0; r < 8; ++r) {
        if ((r & cm) && !(r & tm)) {
          const int r1 = r | tm;
          float t = re[r]; re[r] = re[r1]; re[r1] = t;
          t = im[r]; im[r] = im[r1]; im[r1] = t;
        }
      }
    }
  }
}

__device__ __forceinline__ void cnot_ring(float (&re)[8], float (&im)[8]) {
  // CNOT(i,(i+1)%8): (cb,tb) = (7-i, 7-((i+1)%8))
  cnot<7,6>(re, im); cnot<6,5>(re, im); cnot<5,4>(re, im); cnot<4,3>(re, im);
  cnot<3,2>(re, im); cnot<2,1>(re, im); cnot<1,0>(re, im); cnot<0,7>(re, im);
  // CNOT(i,(i+2)%8)
  cnot<7,5>(re, im); cnot<6,4>(re, im); cnot<5,3>(re, im); cnot<4,2>(re, im);
  cnot<3,1>(re, im); cnot<2,0>(re, im); cnot<1,7>(re, im); cnot<0,6>(re, im);
}

template<int W>
__device__ __forceinline__ void emb_ry(float (&re)[8], float (&im)[8], const float* a) {
  float s, c;
  __sincosf(0.5f * a[W], &s, &c);
  g_ry<7 - W>(re, im, c, s);
  if constexpr (W < 7) emb_ry<W + 1>(re, im, a);
}

template<int W>
__device__ __forceinline__ void emb_rz(float (&re)[8], float (&im)[8], const float* a) {
  float s, c;
  __sincosf(0.5f * a[W], &s, &c);
  g_rz<7 - W>(re, im, c, s);
  if constexpr (W < 7) emb_rz<W + 1>(re, im, a);
}

template<int W>
__device__ __forceinline__ void layer_rot(float (&re)[8], float (&im)[8],
                                          const float* cq, const float* sq) {
  g_rx<7 - W>(re, im, cq[3 * W + 0], sq[3 * W + 0]);
  g_ry<7 - W>(re, im, cq[3 * W + 1], sq[3 * W + 1]);
  g_rz<7 - W>(re, im, cq[3 * W + 2], sq[3 * W + 2]);
  if constexpr (W < 7) layer_rot<W + 1>(re, im, cq, sq);
}

// ---------------------------------------------------------------------------
// Kernel 1: z = x@Wc + bc, ang[.., 0:8] = atan(z), ang[.., 8:16] = atan(z^2)
// via V_WMMA_F32_16X16X4. Wc is staged once per block into LDS, transposed
// and zero-padded to 16x64, so B fragments are branchless ds_load_b64.
// ---------------------------------------------------------------------------
__global__ void __launch_bounds__(256) embed_gemm(const float* __restrict__ x,
                                                  const float* __restrict__ Wc,
                                                  const float* __restrict__ bc,
                                                  float* __restrict__ ang) {
  __shared__ float lWcT[16][64];                 // [n][k], n>=8 zero
  const int tid = threadIdx.x;
#pragma unroll
  for (int idx = 0; idx < 1024; idx += 256) {
    const int i = idx + tid;
    const int n = i >> 6, k = i & 63;
    lWcT[n][k] = (n < WIRES) ? Wc[(size_t)k * WIRES + n] : 0.f;
  }
  __syncthreads();

  const int wave = tid >> 5;
  const int lane = tid & 31;
  const int tile = blockIdx.x * 8 + wave;        // 4096 tiles
  const int row0 = tile * 16;
  const int half = lane >> 4;                    // K-pair selector
  const int mr   = lane & 15;                    // M (A) / N (B,D) index
  const float* xr = x + (size_t)(row0 + mr) * FEAT;
  const float* wr = &lWcT[mr][0];

  v8f acc = {0.f, 0.f, 0.f, 0.f, 0.f, 0.f, 0.f, 0.f};
#pragma unroll
  for (int kk = 0; kk < 16; ++kk) {
    const int k0 = kk * 4 + half * 2;
    v2f a = *(const v2f*)(xr + k0);              // global_load_b64
    v2f b = *(const v2f*)(wr + k0);              // ds_load_b64
    acc = __builtin_amdgcn_wmma_f32_16x16x4_f32(false, a, false, b,
                                                (short)0, acc, false, false);
  }
  if (mr < WIRES) {
    const float bb = bc[mr];
#pragma unroll
    for (int j = 0; j < 8; ++j) {
      const int row = row0 + j + half * 8;       // D layout: vgpr j -> M = j (+8 hi half)
      const float z = acc[j] + bb;
      const size_t o = (size_t)row * 16;
      ang[o + mr]     = atanf(z);
      ang[o + 8 + mr] = atanf(z * z);
    }
  }
}

// ---------------------------------------------------------------------------
// Kernel 2: the sequential circuit scan. One wave32 per batch element;
// statevector fully register-resident (16 VGPRs), shfl_xor for low-bit gates.
// ---------------------------------------------------------------------------
__global__ void __launch_bounds__(256) qrnn_scan(const float* __restrict__ ang,
                                                 const float* __restrict__ qw,
                                                 float* __restrict__ yv) {
  __shared__ float cq[72];
  __shared__ float sq[72];
  const int tid = threadIdx.x;
  if (tid < 72) {
    float s, c;
    __sincosf(0.5f * qw[tid], &s, &c);
    cq[tid] = c; sq[tid] = s;
  }
  __syncthreads();

  const int wave = tid >> 5;
  const int lane = tid & 31;
  const int b    = blockIdx.x * 8 + wave;

  float re[8], im[8];
#pragma unroll
  for (int r = 0; r < 8; ++r) { re[r] = 0.f; im[r] = 0.f; }
  if (lane == 0) re[0] = 1.f;                    // |0...0>

  for (int t = 0; t < TSTEPS; ++t) {
    const size_t rt = (size_t)b * TSTEPS + t;
    const float* ap = ang + rt * 16;             // one 64B line: [ay0..7, az0..7]
    emb_ry<0>(re, im, ap);
    emb_rz<0>(re, im, ap + 8);
#pragma unroll 1
    for (int l = 0; l < 3; ++l) {
      cnot_ring(re, im);
      layer_rot<0>(re, im, cq + 24 * l, sq + 24 * l);
    }
    // <Z_w> : + for bit(7-w)==0, - for ==1
    float p[8], ptot = 0.f;
#pragma unroll
    for (int r = 0; r < 8; ++r) { p[r] = re[r] * re[r] + im[r] * im[r]; ptot += p[r]; }
    float ev[8];
    ev[0] = ev[1] = ev[2] = 0.f;
#pragma unroll
    for (int r = 0; r < 8; ++r) {                // reg bits: wire0->bit7(m=4) etc.
      ev[0] += (r & 4) ? -p[r] : p[r];
      ev[1] += (r & 2) ? -p[r] : p[r];
      ev[2] += (r & 1) ? -p[r] : p[r];
    }
    ev[3] = (lane & 16) ? -ptot : ptot;          // lane bits: wire3->bit4 ... wire7->bit0
    ev[4] = (lane & 8)  ? -ptot : ptot;
    ev[5] = (lane & 4)  ? -ptot : ptot;
    ev[6] = (lane & 2)  ? -ptot : ptot;
    ev[7] = (lane & 1)  ? -ptot : ptot;
#pragma unroll
    for (int off = 16; off >= 1; off >>= 1) {
#pragma unroll
      for (int w = 0; w < 8; ++w) ev[w] += __shfl_xor(ev[w], off, 32);
    }
    if (lane == 0) {
#pragma unroll
      for (int w = 0; w < 8; ++w) yv[rt * WIRES + w] = ev[w];
    }
  }
}

// ---------------------------------------------------------------------------
// Kernel 3: out = y@Wo + bo via V_WMMA_F32_16X16X4 (K=8 -> 2 WMMA per tile)
// One wave per 16x16 output tile of the 65536x128 result.
// ---------------------------------------------------------------------------
__global__ void __launch_bounds__(256) out_gemm(const float* __restrict__ yv,
                                                const float* __restrict__ Wo,
                                                const float* __restrict__ bo,
                                                float* __restrict__ out) {
  const int tid  = threadIdx.x;
  const int wave = tid >> 5;
  const int lane = tid & 31;
  const int tile = blockIdx.x * 8 + wave;        // 32768 tiles
  const int mt = tile >> 3, nt = tile & 7;
  const int row0 = mt * 16, col0 = nt * 16;
  const int half = lane >> 4;
  const int mr   = lane & 15;
  const float* yr = yv + (size_t)(row0 + mr) * WIRES;

  v8f acc = {0.f, 0.f, 0.f, 0.f, 0.f, 0.f, 0.f, 0.f};
#pragma unroll
  for (int kk = 0; kk < 2; ++kk) {
    const int k0 = kk * 4 + half * 2;
    v2f a = *(const v2f*)(yr + k0);
    v2f b;
    b.x = Wo[(size_t)k0 * UNITS + col0 + mr];
    b.y = Wo[(size_t)(k0 + 1) * UNITS + col0 + mr];
    acc = __builtin_amdgcn_wmma_f32_16x16x4_f32(false, a, false, b,
                                                (short)0, acc, false, false);
  }
  const float bb = bo[col0 + mr];
#pragma unroll
  for (int j = 0; j < 8; ++j) {
    const int row = row0 + j + half * 8;
    out[(size_t)row * UNITS + col0 + mr] = acc[j] + bb;
  }
}

// ---------------------------------------------------------------------------
extern "C" void kernel_launch(void* const* d_in, const int* in_sizes, int n_in,
                              void* d_out, int out_size, void* d_ws, size_t ws_size,
                              hipStream_t stream) {
  const float* x  = (const float*)d_in[0];   // (256,256,64)
  const float* Wc = (const float*)d_in[1];   // (64,8)
  const float* bc = (const float*)d_in[2];   // (8,)
  const float* qw = (const float*)d_in[3];   // (3,24)
  const float* Wo = (const float*)d_in[4];   // (8,128)
  const float* bo = (const float*)d_in[5];   // (128,)
  float* out = (float*)d_out;                // (256,256,128)

  float* ws  = (float*)d_ws;
  float* ang = ws;                           // BT*16 floats = 4 MB (ay|az interleaved)
  float* yv  = ws + (size_t)BT * 16;         // BT*8 floats = 2 MB

  embed_gemm<<<dim3(BT / 16 / 8), dim3(256), 0, stream>>>(x, Wc, bc, ang);
  qrnn_scan<<<dim3(BATCH / 8), dim3(256), 0, stream>>>(ang, qw, yv);
  out_gemm<<<dim3((BT / 16) * (UNITS / 16) / 8), dim3(256), 0, stream>>>(yv, Wo, bo, out);
}